// S4Embedding_19877108646485
// MI455X (gfx1250) — compile-verified
//
#include <hip/hip_runtime.h>

typedef float v2f __attribute__((ext_vector_type(2)));
typedef float v8f __attribute__((ext_vector_type(8)));

#define TOTAL_TOKENS (8 * 2048)   // 16384
#define DPROJ 1024
#define EMB_SCALE 32.0f           // sqrt(1024)

// ws layout: [0..3] int counters | 4 * TOTAL_TOKENS int lists | 512-float dump
#define WS_DUMP_OFF (4 + 4 * TOTAL_TOKENS)   // in ints/floats

// ---------------------------------------------------------------------------
// Phase 1: bucket tokens by cluster into per-cluster lists in workspace.
// ---------------------------------------------------------------------------
__global__ __launch_bounds__(256) void partition_kernel(
    const int* __restrict__ ids, int* __restrict__ counts, int* __restrict__ lists) {
  int t = blockIdx.x * blockDim.x + threadIdx.x;
  if (t >= TOTAL_TOKENS) return;
  int id = ids[t];
  int c = (id < 20000) ? 0 : (id < 40000) ? 1 : (id < 200000) ? 2 : 3;
  int p = atomicAdd(&counts[c], 1);
  lists[c * TOTAL_TOKENS + p] = t;
}

// ---------------------------------------------------------------------------
// Phase 2: grouped GEMM per cluster using V_WMMA_F32_16X16X4_F32.
// grid: x = N-block (4 x 256 cols), y = M-tile (16 tokens), z = cluster.
// Each wave computes a 16(M) x 32(N) tile as two accumulators sharing one
// A stream (gathered embedding rows); K-loop in steps of 4.
// Branchless scatter epilogue: padded rows of a partial tile are redirected
// to a dump area via address select (no EXEC manipulation after early-exit).
// ---------------------------------------------------------------------------
__global__ __launch_bounds__(256) void adaptive_emb_gemm(
    const int* __restrict__ ids,
    const float* __restrict__ emb0, const float* __restrict__ proj0,
    const float* __restrict__ emb1, const float* __restrict__ proj1,
    const float* __restrict__ emb2, const float* __restrict__ proj2,
    const float* __restrict__ emb3, const float* __restrict__ proj3,
    const int* __restrict__ counts, const int* __restrict__ lists,
    float* __restrict__ dump, float* __restrict__ out) {
  const int c = blockIdx.z;
  const int cnt = counts[c];                 // scalar load, uniform
  const int mBase = blockIdx.y * 16;
  if (mBase >= cnt) return;                  // uniform -> EXEC stays all-1s below

  const float* emb; const float* proj; int d, lo, rows;
  switch (c) {
    case 0:  emb = emb0; proj = proj0; d = 1024; lo = 0;      rows = 20000;  break;
    case 1:  emb = emb1; proj = proj1; d = 256;  lo = 20000;  rows = 20000;  break;
    case 2:  emb = emb2; proj = proj2; d = 64;   lo = 40000;  rows = 160000; break;
    default: emb = emb3; proj = proj3; d = 16;   lo = 200000; rows = 67735;  break;
  }

  const int lane  = threadIdx.x & 31;
  const int wave  = threadIdx.x >> 5;
  const int laneM = lane & 15;       // row (A) / col (B) index within a 16-tile
  const int hi    = lane >> 4;       // 0 -> K pair {0,1}; 1 -> K pair {2,3}
  const int n0    = blockIdx.x * 256 + wave * 32;   // this wave: cols n0..n0+31

  // A-row source: this lane's token (lanes 16-31 mirror rows of lanes 0-15).
  const int  m      = mBase + laneM;
  const bool mValid = (m < cnt);
  const int  pos    = lists[c * TOTAL_TOKENS + (mValid ? m : mBase)];
  int idx = ids[pos] - lo;
  idx = idx < 0 ? 0 : (idx >= rows ? rows - 1 : idx);   // clamp (pad lanes only)

  const float* aRow  = emb  + (size_t)idx * d;
  const float* bRow0 = proj + (size_t)(n0 + laneM) * d;        // B[e][n] = proj[n*d + e]
  const float* bRow1 = proj + (size_t)(n0 + 16 + laneM) * d;

  v8f acc0 = {};
  v8f acc1 = {};
#pragma unroll 4
  for (int k = 0; k < d; k += 4) {
    v2f a  = *(const v2f*)(aRow  + k + 2 * hi);
    v2f b0 = *(const v2f*)(bRow0 + k + 2 * hi);
    v2f b1 = *(const v2f*)(bRow1 + k + 2 * hi);
    acc0 = __builtin_amdgcn_wmma_f32_16x16x4_f32(false, a, false, b0,
                                                 (short)0, acc0, false, false);
    acc1 = __builtin_amdgcn_wmma_f32_16x16x4_f32(false, a, false, b1,
                                                 (short)0, acc1, false, false);
  }

  // Branchless scatter epilogue. VGPR r holds row (r + 8*hi), col n0+laneM
  // (acc0) and n0+16+laneM (acc1). Row r is valid iff r < vr (per half-wave).
  const int listBase = c * TOTAL_TOKENS + mBase + 8 * hi;
  const int vr = cnt - mBase - 8 * hi;        // rows valid while r < vr
  float* dumpP = dump + threadIdx.x;          // write-only scratch for pad rows
#pragma unroll
  for (int r = 0; r < 8; ++r) {
    int rpos = lists[listBase + r] & (TOTAL_TOKENS - 1);  // mask stale/pad reads
    float* p = (r < vr) ? (out + (size_t)rpos * DPROJ + n0 + laneM) : dumpP;
    p[0]  = acc0[r] * EMB_SCALE;
    p[16] = acc1[r] * EMB_SCALE;
  }
}

extern "C" void kernel_launch(void* const* d_in, const int* in_sizes, int n_in,
                              void* d_out, int out_size, void* d_ws, size_t ws_size,
                              hipStream_t stream) {
  // setup_inputs order: input_ids, emb0, proj0, emb1, proj1, emb2, proj2, emb3, proj3
  const int*   ids   = (const int*)d_in[0];
  const float* emb0  = (const float*)d_in[1];
  const float* proj0 = (const float*)d_in[2];
  const float* emb1  = (const float*)d_in[3];
  const float* proj1 = (const float*)d_in[4];
  const float* emb2  = (const float*)d_in[5];
  const float* proj2 = (const float*)d_in[6];
  const float* emb3  = (const float*)d_in[7];
  const float* proj3 = (const float*)d_in[8];
  float* out = (float*)d_out;

  int*   counts = (int*)d_ws;
  int*   lists  = counts + 4;                 // 4 * TOTAL_TOKENS ints
  float* dump   = (float*)d_ws + WS_DUMP_OFF; // 512-float write-only scratch

  hipMemsetAsync(counts, 0, 4 * sizeof(int), stream);

  partition_kernel<<<(TOTAL_TOKENS + 255) / 256, 256, 0, stream>>>(ids, counts, lists);

  // 4 N-blocks (256 cols each) x 1024 M-tiles (max 16384 tokens / 16) x 4 clusters.
  dim3 grid(4, TOTAL_TOKENS / 16, 4);
  adaptive_emb_gemm<<<grid, 256, 0, stream>>>(
      ids, emb0, proj0, emb1, proj1, emb2, proj2, emb3, proj3, counts, lists,
      dump, out);
}